// RegionProposalNetwork_fpn_35416300323737
// MI455X (gfx1250) — compile-verified
//
#include <hip/hip_runtime.h>
#include <stdint.h>

#define BB        16
#define PRE_NMS   2000
#define POST_NMS  1000
#define BINS      4096      // 12-bit radix on order-preserving key
#define CAP       4096      // candidate buffer per image
#define HCHUNK    32768     // floats per block per streaming pass
#define TILE      8192      // floats per LDS staging tile (32KB)
#define NMS_THR   0.7f
#define MIN_ANGLE 10.0f
#define IMG_RANGE 360.0f
#define LOG_W_MAX 4.135f

// order-preserving float -> u32 key (larger key == larger float)
__device__ __forceinline__ uint32_t f2key(float f) {
  uint32_t u = __float_as_uint(f);
  return (u & 0x80000000u) ? ~u : (u | 0x80000000u);
}

// CDNA5 async global->LDS B128 copy (per-lane, tracked on ASYNCcnt)
__device__ __forceinline__ void async_b128(const float* g, float* l) {
  uint32_t laddr = (uint32_t)(uintptr_t)l;
  uint64_t gaddr = (uint64_t)(uintptr_t)g;
  asm volatile("global_load_async_to_lds_b128 %0, %1, off"
               :: "v"(laddr), "v"(gaddr) : "memory");
}
__device__ __forceinline__ void wait_async0() {
  asm volatile("s_wait_asynccnt 0" ::: "memory");
}

// ---------------------------------------------------------------------------
// Pass 1: per-image 4096-bin histogram of key>>20, streamed via CDNA5 async
// global->LDS loads, LDS atomics, merged to global.
// ---------------------------------------------------------------------------
__global__ __launch_bounds__(256) void rpn_hist(const float* __restrict__ obj,
                                                uint32_t* __restrict__ hist,
                                                int N) {
  __shared__ uint32_t h[BINS];
  __shared__ float tile[TILE];
  const int t = threadIdx.x;
  const int b = blockIdx.y;

  for (int i = t; i < BINS; i += 256) h[i] = 0u;
  __syncthreads();

  const int chunk0 = blockIdx.x * HCHUNK;
  int remain = N - chunk0;
  if (remain > HCHUNK) remain = HCHUNK;
  const float* src = obj + (size_t)b * N + chunk0;

  for (int off = 0; off < remain; off += TILE) {
    int cnt = remain - off;
    if (cnt > TILE) cnt = TILE;
    int nvec = cnt >> 2;  // 16B units
    for (int i = t; i < nvec; i += 256)
      async_b128(src + off + i * 4, &tile[i * 4]);
    for (int i = (nvec << 2) + t; i < cnt; i += 256)  // scalar tail
      tile[i] = src[off + i];
    wait_async0();
    __syncthreads();

    for (int i = t; i < cnt; i += 256)
      atomicAdd(&h[f2key(tile[i]) >> 20], 1u);
    __syncthreads();
  }

  uint32_t* gh = hist + (size_t)b * BINS;
  for (int i = t; i < BINS; i += 256) {
    uint32_t c = h[i];
    if (c) atomicAdd(&gh[i], c);
  }
}

// ---------------------------------------------------------------------------
// Pass 2: per-image threshold bin: count(bins > tb) < PRE_NMS <= count(>= tb)
// ---------------------------------------------------------------------------
__global__ void rpn_thresh(const uint32_t* __restrict__ hist,
                           uint32_t* __restrict__ info) {
  int b = threadIdx.x;
  if (b >= BB) return;
  const uint32_t* h = hist + (size_t)b * BINS;
  uint32_t cum = 0, tb = 0;
  for (int i = BINS - 1; i >= 0; --i) {
    uint32_t c = h[i];
    if (cum + c >= PRE_NMS) { tb = (uint32_t)i; break; }
    cum += c;
  }
  info[b * 2 + 0] = tb;
  info[b * 2 + 1] = cum;
}

// ---------------------------------------------------------------------------
// Pass 3: second stream (L2-resident: 64MB obj fits 192MB L2) via the same
// async global->LDS path; compact all (key, idx) with bin >= tb.
// ---------------------------------------------------------------------------
__global__ __launch_bounds__(256) void rpn_collect(const float* __restrict__ obj,
                                                   const uint32_t* __restrict__ info,
                                                   uint32_t* __restrict__ cnt,
                                                   uint32_t* __restrict__ ckey,
                                                   uint32_t* __restrict__ cidx,
                                                   int N) {
  __shared__ float tile[TILE];
  const int t = threadIdx.x;
  const int b = blockIdx.y;
  const uint32_t tb = info[b * 2];

  const int chunk0 = blockIdx.x * HCHUNK;
  int remain = N - chunk0;
  if (remain > HCHUNK) remain = HCHUNK;
  const float* src = obj + (size_t)b * N + chunk0;

  for (int off = 0; off < remain; off += TILE) {
    int c = remain - off;
    if (c > TILE) c = TILE;
    int nvec = c >> 2;
    for (int i = t; i < nvec; i += 256)
      async_b128(src + off + i * 4, &tile[i * 4]);
    for (int i = (nvec << 2) + t; i < c; i += 256)
      tile[i] = src[off + i];
    wait_async0();
    __syncthreads();

    for (int i = t; i < c; i += 256) {
      uint32_t k = f2key(tile[i]);
      if ((k >> 20) >= tb) {
        uint32_t slot = atomicAdd(&cnt[b], 1u);
        if (slot < CAP) {
          ckey[(size_t)b * CAP + slot] = k;
          cidx[(size_t)b * CAP + slot] = (uint32_t)(chunk0 + off + i);
        }
      }
    }
    __syncthreads();
  }
}

// ---------------------------------------------------------------------------
// Pass 4 (1 block / image): bitonic sort desc -> decode -> greedy NMS -> scan
// -> compact first 1000 survivors (score order preserved => post top-k).
// ---------------------------------------------------------------------------
__global__ __launch_bounds__(1024) void rpn_final(const float* __restrict__ delta,
                                                  const float* __restrict__ anchor,
                                                  const uint32_t* __restrict__ cnt,
                                                  const uint32_t* __restrict__ ckey,
                                                  const uint32_t* __restrict__ cidx,
                                                  float* __restrict__ out,
                                                  int N) {
  __shared__ uint32_t skey[CAP];
  __shared__ uint32_t sidx[CAP];
  __shared__ float slo[2048], shi[2048], sar[2048];
  __shared__ int skeep[2048];
  __shared__ int sscan[2048];
  const int b = blockIdx.x, t = threadIdx.x;

  uint32_t m = cnt[b];
  if (m > CAP) m = CAP;
  for (int i = t; i < CAP; i += 1024) {
    bool v = (uint32_t)i < m;
    skey[i] = v ? ckey[(size_t)b * CAP + i] : 0u;
    sidx[i] = v ? cidx[(size_t)b * CAP + i] : 0u;
  }
  __syncthreads();

  // bitonic sort, descending by key
  for (int size = 2; size <= CAP; size <<= 1) {
    for (int stride = size >> 1; stride > 0; stride >>= 1) {
      for (int i = t; i < CAP; i += 1024) {
        int j = i ^ stride;
        if (j > i) {
          bool up = ((i & size) == 0);
          uint32_t ki = skey[i], kj = skey[j];
          bool sw = up ? (ki < kj) : (ki > kj);
          if (sw) {
            skey[i] = kj; skey[j] = ki;
            uint32_t ti = sidx[i]; sidx[i] = sidx[j]; sidx[j] = ti;
          }
        }
      }
      __syncthreads();
    }
  }

  // decode + clip + min-width filter for top PRE_NMS (sorted desc)
  for (int i = t; i < 2048; i += 1024) {
    float l = 0.f, h = 0.f;
    int v = 0;
    if (i < PRE_NMS && (uint32_t)i < m) {
      size_t base = ((size_t)b * N + sidx[i]) * 2;
      float a0 = anchor[base], a1 = anchor[base + 1];
      float d0 = delta[base],  d1 = delta[base + 1];
      float aw = a1 - a0;
      float ac = a0 + 0.5f * aw;
      float dw = fminf(d1, LOG_W_MAX);
      float pc = d0 * aw + ac;
      float pw = expf(dw) * aw;
      l = pc - 0.5f * pw;
      h = pc + 0.5f * pw;
      l = fminf(fmaxf(l, 0.f), IMG_RANGE);
      h = fminf(fmaxf(h, 0.f), IMG_RANGE);
      v = ((h - l) >= MIN_ANGLE) ? 1 : 0;
    }
    slo[i] = l; shi[i] = h; sar[i] = h - l; skeep[i] = v;
  }
  __syncthreads();

  // greedy interval NMS (exact sequential-greedy semantics)
  for (int i = 0; i < PRE_NMS; ++i) {
    if (skeep[i]) {  // uniform: all threads read same post-barrier value
      float li = slo[i], hi_ = shi[i], ai = sar[i];
      for (int j = i + 1 + t; j < PRE_NMS; j += 1024) {
        if (skeep[j]) {
          float inter = fminf(hi_, shi[j]) - fmaxf(li, slo[j]);
          inter = fmaxf(inter, 0.f);
          float iou = inter / fmaxf(ai + sar[j] - inter, 1e-8f);
          if (iou > NMS_THR) skeep[j] = 0;
        }
      }
    }
    __syncthreads();
  }

  // inclusive scan over keep flags (Hillis-Steele, 2048 elems)
  for (int i = t; i < 2048; i += 1024) sscan[i] = (i < PRE_NMS) ? skeep[i] : 0;
  __syncthreads();
  for (int off = 1; off < 2048; off <<= 1) {
    int i1 = t, i2 = t + 1024;
    int a = (i1 >= off) ? sscan[i1 - off] : 0;
    int c = (i2 >= off) ? sscan[i2 - off] : 0;
    __syncthreads();
    sscan[i1] += a;
    sscan[i2] += c;
    __syncthreads();
  }

  // survivors remain score-sorted -> first POST_NMS kept are the output
  for (int i = t; i < PRE_NMS; i += 1024) {
    if (skeep[i]) {
      int r = sscan[i] - 1;
      if (r < POST_NMS) {
        size_t o = ((size_t)b * POST_NMS + r) * 2;
        out[o] = slo[i];
        out[o + 1] = shi[i];
      }
    }
  }
}

// ---------------------------------------------------------------------------
extern "C" void kernel_launch(void* const* d_in, const int* in_sizes, int n_in,
                              void* d_out, int out_size, void* d_ws, size_t ws_size,
                              hipStream_t stream) {
  const float* obj    = (const float*)d_in[0];
  const float* delta  = (const float*)d_in[1];
  const float* anchor = (const float*)d_in[2];
  const int B = BB;
  const int N = in_sizes[0] / B;

  uint8_t* ws = (uint8_t*)d_ws;
  size_t off_hist = 0;
  size_t off_info = off_hist + (size_t)B * BINS * 4;   // B*2 u32
  size_t off_cnt  = off_info + 256;                    // B u32
  size_t off_ckey = off_cnt + 256;
  size_t off_cidx = off_ckey + (size_t)B * CAP * 4;

  uint32_t* hist = (uint32_t*)(ws + off_hist);
  uint32_t* info = (uint32_t*)(ws + off_info);
  uint32_t* cnt  = (uint32_t*)(ws + off_cnt);
  uint32_t* ckey = (uint32_t*)(ws + off_ckey);
  uint32_t* cidx = (uint32_t*)(ws + off_cidx);

  // zero histogram/info/counters every call (graph-replay safe), zero output
  hipMemsetAsync(ws, 0, off_ckey, stream);
  hipMemsetAsync(d_out, 0, (size_t)out_size * sizeof(float), stream);

  const int nchunks = (N + HCHUNK - 1) / HCHUNK;
  rpn_hist<<<dim3(nchunks, B), 256, 0, stream>>>(obj, hist, N);
  rpn_thresh<<<1, 32, 0, stream>>>(hist, info);
  rpn_collect<<<dim3(nchunks, B), 256, 0, stream>>>(obj, info, cnt, ckey, cidx, N);
  rpn_final<<<B, 1024, 0, stream>>>(delta, anchor, cnt, ckey, cidx, (float*)d_out, N);
}